// S4_11003706212529
// MI455X (gfx1250) — compile-verified
//
#include <hip/hip_runtime.h>

// S4 FFT convolution for MI455X (gfx1250, wave32).
// FFT(8192) = 16 x 16 x 32 four-step: two DFT-16 stages as complex matmuls on
// V_WMMA_F32_16X16X4_F32, one 32-point FFT fully in registers, all twiddles
// fused.  K spectrum row prefetched to LDS via the Tensor Data Mover.

typedef float v2f __attribute__((ext_vector_type(2)));
typedef float v8f __attribute__((ext_vector_type(8)));
typedef unsigned int u32x4 __attribute__((ext_vector_type(4)));
typedef int i32x4 __attribute__((ext_vector_type(4)));
typedef int i32x8 __attribute__((ext_vector_type(8)));

#define WG 256
#define PI2 6.28318530717958647692f

// [16][16][32] layout for 8192-pt phase (c padded 32->36 for b128 alignment)
#define AS8 576
#define BS8 36
// [16][16][16] layout for 4096-pt phase (c padded 16->20)
#define AS4 320
#define BS4 20

#if __has_builtin(__builtin_amdgcn_tensor_load_to_lds)
#define S4_HAS_TDM 1
#endif

__device__ __forceinline__ unsigned bitrev(unsigned x, int bits) {
    return __brev(x) >> (32 - bits);
}

// W32 twiddle tables: cos/sin(2*pi*t/32), t = 0..15 (W16 entries at even idx)
__device__ __constant__ float TWC32[16] = {
    1.0f,  0.98078528f,  0.92387953f,  0.83146961f,  0.70710678f,  0.55557023f,
    0.38268343f,  0.19509032f,  0.0f, -0.19509032f, -0.38268343f, -0.55557023f,
   -0.70710678f, -0.83146961f, -0.92387953f, -0.98078528f};
__device__ __constant__ float TWS32[16] = {
    0.0f,  0.19509032f,  0.38268343f,  0.55557023f,  0.70710678f,  0.83146961f,
    0.92387953f,  0.98078528f,  1.0f,  0.98078528f,  0.92387953f,  0.83146961f,
    0.70710678f,  0.55557023f,  0.38268343f,  0.19509032f};

// ---------------------------------------------------------------------------
// Fully register-resident radix-2 FFTs of length C = 1<<LOGC (C = 16 or 32).
// DIF: natural in -> bit-reversed out.  DIT: bit-reversed in -> natural out.
// ---------------------------------------------------------------------------
template <int LOGC>
__device__ __forceinline__ void reg_fft_dif(float* xr, float* xi, float dsign) {
    const int C = 1 << LOGC;
#pragma unroll
    for (int s = LOGC - 1; s >= 0; --s) {
#pragma unroll
        for (int b = 0; b < (C >> 1); ++b) {
            const int span = 1 << s;
            const int off = b & (span - 1);
            const int grp = b >> s;
            const int i0 = (grp << (s + 1)) + off;
            const int i1 = i0 + span;
            const int tw = (off << (LOGC - 1 - s)) << (5 - LOGC);
            float wr = TWC32[tw], wi = dsign * TWS32[tw];
            float ar = xr[i0], ai = xi[i0];
            float cr = xr[i1], ci = xi[i1];
            xr[i0] = ar + cr;  xi[i0] = ai + ci;
            float tr = ar - cr, ti = ai - ci;
            xr[i1] = tr * wr - ti * wi;
            xi[i1] = tr * wi + ti * wr;
        }
    }
}

template <int LOGC>
__device__ __forceinline__ void reg_fft_dit(float* xr, float* xi, float dsign) {
    const int C = 1 << LOGC;
#pragma unroll
    for (int s = 0; s < LOGC; ++s) {
#pragma unroll
        for (int b = 0; b < (C >> 1); ++b) {
            const int span = 1 << s;
            const int off = b & (span - 1);
            const int grp = b >> s;
            const int i0 = (grp << (s + 1)) + off;
            const int i1 = i0 + span;
            const int tw = (off << (LOGC - 1 - s)) << (5 - LOGC);
            float wr = TWC32[tw], wi = dsign * TWS32[tw];
            float cr = xr[i1], ci = xi[i1];
            float tr = cr * wr - ci * wi;
            float ti = cr * wi + ci * wr;
            float ar = xr[i0], ai = xi[i0];
            xr[i1] = ar - tr;  xi[i1] = ai - ti;
            xr[i0] = ar + tr;  xi[i0] = ai + ti;
        }
    }
}

// ---------------------------------------------------------------------------
// DFT-16 over one 16-long axis for all 16*C columns, as complex 16x16 matmul
// via chained V_WMMA_F32_16X16X4_F32 (K chunked 4x4).  Element (r, colA, c)
// lives at r*RS + colA*CS + c.  In-place: each wave owns disjoint 16-column
// tiles.  Fused output twiddles:
//   mode 0: none
//   mode 1: exp(dsign*i*2pi * colA*(br(c) + row<<LOGC) * invN)   (fwd outer tw)
//   mode 2: exp(dsign*i*2pi * row *(br(c) + colA<<LOGC) * invN)  (inv outer tw)
//   mode 3: exp(dsign*i*2pi * row * br(c) * invN)                (inner tw)
// ---------------------------------------------------------------------------
__device__ void dft16_wmma(float* GR, float* GI, int RS, int CS, int LOGC,
                           int mode, float dsign, float invN) {
    const int tid  = threadIdx.x;
    const int lane = tid & 31;
    const int wave = tid >> 5;
    const int hi   = (lane >= 16) ? 1 : 0;
    const int M    = lane & 15;
    // A = W16^(dsign*M*K); 16x4 A layout: lanes0-15 K={4kc,4kc+1}, lanes16-31 +2
    v2f Ar[4], Ai[4], AiN[4];
#pragma unroll
    for (int kc = 0; kc < 4; ++kc) {
        int k0 = kc * 4 + hi * 2;
        int t0 = ((M * k0) & 15) << 1;        // index into W32 table (W16 = even)
        int t1 = ((M * (k0 + 1)) & 15) << 1;
        Ar[kc].x = TWC32[t0];          Ar[kc].y = TWC32[t1];
        Ai[kc].x = dsign * TWS32[t0];  Ai[kc].y = dsign * TWS32[t1];
        AiN[kc].x = -Ai[kc].x;         AiN[kc].y = -Ai[kc].y;
    }
    const int C   = 1 << LOGC;
    const int tpw = ((16 << LOGC) >> 4) >> 3;   // tiles per wave (8 waves)
    __syncthreads();
    for (int t = 0; t < tpw; ++t) {
        int q    = ((wave * tpw + t) << 4) + M;
        int colA = q >> LOGC;
        int c    = q & (C - 1);
        int cb   = colA * CS + c;
        v2f Br[4], Bi[4];
#pragma unroll
        for (int kc = 0; kc < 4; ++kc) {
            int r0 = kc * 4 + hi * 2;
            int p0 = r0 * RS + cb;
            Br[kc].x = GR[p0];  Br[kc].y = GR[p0 + RS];
            Bi[kc].x = GI[p0];  Bi[kc].y = GI[p0 + RS];
        }
        v8f yr = {0.f,0.f,0.f,0.f,0.f,0.f,0.f,0.f};
        v8f yi = {0.f,0.f,0.f,0.f,0.f,0.f,0.f,0.f};
#pragma unroll
        for (int kc = 0; kc < 4; ++kc)
            yr = __builtin_amdgcn_wmma_f32_16x16x4_f32(false, Ar[kc],  false, Br[kc], (short)0, yr, false, false);
#pragma unroll
        for (int kc = 0; kc < 4; ++kc)
            yr = __builtin_amdgcn_wmma_f32_16x16x4_f32(false, AiN[kc], false, Bi[kc], (short)0, yr, false, false);
#pragma unroll
        for (int kc = 0; kc < 4; ++kc)
            yi = __builtin_amdgcn_wmma_f32_16x16x4_f32(false, Ar[kc],  false, Bi[kc], (short)0, yi, false, false);
#pragma unroll
        for (int kc = 0; kc < 4; ++kc)
            yi = __builtin_amdgcn_wmma_f32_16x16x4_f32(false, Ai[kc],  false, Br[kc], (short)0, yi, false, false);
        int k2v = (int)bitrev((unsigned)c, LOGC);
#pragma unroll
        for (int g = 0; g < 8; ++g) {
            int row = g + hi * 8;
            int p = row * RS + cb;
            float orv = yr[g], oiv = yi[g];
            if (mode != 0) {
                int f = (mode == 1) ? colA * (k2v + (row << LOGC))
                      : (mode == 2) ? row * (k2v + (colA << LOGC))
                                    : row * k2v;
                float ang = dsign * PI2 * (float)f * invN;
                float wr, wi;
                __sincosf(ang, &wi, &wr);
                float tmp = orv * wr - oiv * wi;
                oiv = orv * wi + oiv * wr;
                orv = tmp;
            }
            GR[p] = orv;
            GI[p] = oiv;
        }
    }
    __syncthreads();
}

// ===========================================================================
// Kernel 1: Cauchy kernel + rank-1 Woodbury -> kf[h][m], m in [0, 2049)
// ===========================================================================
__global__ void s4_cauchy_kernel(const float* __restrict__ log_dt,
                                 const float* __restrict__ ivr,
                                 const float* __restrict__ wim,
                                 const float* __restrict__ Bp,
                                 const float* __restrict__ Pp,
                                 const float* __restrict__ Cp,
                                 float2* __restrict__ kf) {
    __shared__ float sw[16][2];
    __shared__ float sv00[16][2], sv01[16][2], sv10[16][2], sv11[16][2];
    __shared__ float sdt;
    const int h = blockIdx.x;
    const int tid = threadIdx.x;
    if (tid == 0) sdt = __expf(log_dt[h]);
    __syncthreads();
    if (tid < 16) {
        int n = tid;
        float dt  = sdt;
        float wr  = -__expf(ivr[h * 16 + n]) * dt;
        float wi  = wim[h * 16 + n] * dt;
        float brr = Bp[(h * 16 + n) * 2 + 0], bri = Bp[(h * 16 + n) * 2 + 1];
        float pr  = Pp[(h * 16 + n) * 2 + 0], pi  = Pp[(h * 16 + n) * 2 + 1];
        float cr  = Cp[(h * 16 + n) * 2 + 0], ci  = Cp[(h * 16 + n) * 2 + 1];
        float qr  = pr, qi = -pi;
        sw[n][0] = wr;  sw[n][1] = wi;
        sv00[n][0] = brr * cr - bri * ci;  sv00[n][1] = brr * ci + bri * cr;
        sv01[n][0] = brr * qr - bri * qi;  sv01[n][1] = brr * qi + bri * qr;
        sv10[n][0] = pr * cr - pi * ci;    sv10[n][1] = pr * ci + pi * cr;
        sv11[n][0] = pr * qr - pi * qi;    sv11[n][1] = pr * qi + pi * qr;
    }
    __syncthreads();
    const float dt = sdt;
    for (int m = tid; m < 2049; m += WG) {
        float ang = -PI2 * (float)m * (1.0f / 4096.0f);
        float owi, owr;
        __sincosf(ang, &owi, &owr);
        float tr = 1.0f + owr, ti = owi;
        float id = 1.0f / (tr * tr + ti * ti);
        float nr = 1.0f - owr, ni = -owi;
        float zr = 2.0f * (nr * tr + ni * ti) * id;
        float zi = 2.0f * (ni * tr - nr * ti) * id;
        float r00r = 0.f, r00i = 0.f, r01r = 0.f, r01i = 0.f;
        float r10r = 0.f, r10i = 0.f, r11r = 0.f, r11i = 0.f;
#pragma unroll
        for (int n = 0; n < 16; ++n) {
            float dr = zr - sw[n][0], di = zi - sw[n][1];
            float idn = 1.0f / (dr * dr + di * di);
            float gr = dr * idn, gi = -di * idn;
            r00r += sv00[n][0] * gr - sv00[n][1] * gi;  r00i += sv00[n][0] * gi + sv00[n][1] * gr;
            r01r += sv01[n][0] * gr - sv01[n][1] * gi;  r01i += sv01[n][0] * gi + sv01[n][1] * gr;
            r10r += sv10[n][0] * gr - sv10[n][1] * gi;  r10i += sv10[n][0] * gi + sv10[n][1] * gr;
            r11r += sv11[n][0] * gr - sv11[n][1] * gi;  r11i += sv11[n][0] * gi + sv11[n][1] * gr;
        }
        r00r *= dt; r00i *= dt; r01r *= dt; r01i *= dt;
        r10r *= dt; r10i *= dt; r11r *= dt; r11i *= dt;
        float ar = 1.0f + r11r, ai = r11i;
        float ia = 1.0f / (ar * ar + ai * ai);
        float numr = r01r * r10r - r01i * r10i;
        float numi = r01r * r10i + r01i * r10r;
        float ur = (numr * ar + numi * ai) * ia;
        float ui = (numi * ar - numr * ai) * ia;
        float kr = r00r - ur, ki = r00i - ui;
        float fr = 2.0f * tr * id, fi = -2.0f * ti * id;
        float2 o;
        o.x = kr * fr - ki * fi;
        o.y = kr * fi + ki * fr;
        kf[(size_t)h * 2049 + m] = o;
    }
}

// ===========================================================================
// Kernel 2: kf -> irfft(4096) -> rfft(8192) -> K (pipeline storage order:
// linear idx = k1*512 + kap1*32 + j  with  k = br5(j) + 32*kap1 + 512*k1)
// ===========================================================================
__global__ void s4_kf_to_K(const float2* __restrict__ kf, float2* __restrict__ K) {
    extern __shared__ float smem[];
    float* XR = smem;           // 9216 floats (also used as [16][16][20] layout)
    float* XI = XR + 9216;
    const int h = blockIdx.x;
    const int tid = threadIdx.x;
    // Hermitian-extend into the inverse pipeline's input layout [k1][kap1][j]
    for (int idx = tid; idx < 4096; idx += WG) {
        int aa = idx >> 8, bq = (idx >> 4) & 15, j = idx & 15;
        int k = (int)bitrev((unsigned)j, 4) + (bq << 4) + (aa << 8);
        float2 v;
        if (k <= 2048) {
            v = kf[(size_t)h * 2049 + k];
        } else {
            v = kf[(size_t)h * 2049 + (4096 - k)];
            v.y = -v.y;
        }
        XR[aa * AS4 + bq * BS4 + j] = v.x;
        XI[aa * AS4 + bq * BS4 + j] = v.y;
    }
    // inverse 4096: DFT16 over k1 (+outer tw), DFT16 over kap1 (+inner tw)
    dft16_wmma(XR, XI, AS4, BS4, 4, 2, +1.0f, 1.0f / 4096.0f);
    dft16_wmma(XR, XI, BS4, AS4, 4, 3, +1.0f, 1.0f / 256.0f);
    // fused: register iFFT-16 (time-domain k), zero-pad, register FFT-32 +
    // W512 twiddle = first stage of forward 8192 (no LDS round-trip between)
    {
        const int a = tid >> 4, b = tid & 15;
        float vr[16], vi[16];
        float4* pr  = (float4*)(XR + a * AS4 + b * BS4);
        float4* pi_ = (float4*)(XI + a * AS4 + b * BS4);
#pragma unroll
        for (int q = 0; q < 4; ++q) {
            float4 r4 = pr[q], i4 = pi_[q];
            vr[4*q+0] = r4.x; vr[4*q+1] = r4.y; vr[4*q+2] = r4.z; vr[4*q+3] = r4.w;
            vi[4*q+0] = i4.x; vi[4*q+1] = i4.y; vi[4*q+2] = i4.z; vi[4*q+3] = i4.w;
        }
        __syncthreads();   // all reads done before layout-B overwrites
        reg_fft_dit<4>(vr, vi, +1.0f);
        float xr[32], xi2[32];
#pragma unroll
        for (int c = 0; c < 16; ++c) {
            xr[c] = vr[c] * (1.0f / 4096.0f);   // real kernel k[n], irfft scale
            xi2[c] = 0.0f;
            xr[c + 16] = 0.0f;  xi2[c + 16] = 0.0f;
        }
        reg_fft_dif<5>(xr, xi2, -1.0f);
#pragma unroll
        for (int j = 0; j < 32; ++j) {          // fused W512^(-m1*kap2)
            int k2 = (int)bitrev((unsigned)j, 5);
            float ang = -PI2 * (float)(b * k2) * (1.0f / 512.0f);
            float wr, wi;
            __sincosf(ang, &wi, &wr);
            float t1 = xr[j] * wr - xi2[j] * wi;
            xi2[j] = xr[j] * wi + xi2[j] * wr;
            xr[j] = t1;
        }
        float4* qr = (float4*)(XR + a * AS8 + b * BS8);
        float4* qi = (float4*)(XI + a * AS8 + b * BS8);
#pragma unroll
        for (int q = 0; q < 8; ++q) {
            qr[q] = make_float4(xr[4*q], xr[4*q+1], xr[4*q+2], xr[4*q+3]);
            qi[q] = make_float4(xi2[4*q], xi2[4*q+1], xi2[4*q+2], xi2[4*q+3]);
        }
    }
    // forward 8192: DFT16 over m1 (+outer tw), DFT16 over n1
    dft16_wmma(XR, XI, BS8, AS8, 5, 1, -1.0f, 1.0f / 8192.0f);
    dft16_wmma(XR, XI, AS8, BS8, 5, 0, -1.0f, 0.0f);
    for (int idx = tid; idx < 8192; idx += WG) {
        int aa = idx >> 9, bq = (idx >> 5) & 15, c = idx & 31;
        float2 v;
        v.x = XR[aa * AS8 + bq * BS8 + c];
        v.y = XI[aa * AS8 + bq * BS8 + c];
        K[(size_t)h * 8192 + idx] = v;
    }
}

// ===========================================================================
// Kernel 3: per (b,h) row: FFT(8192) -> *K[h] (TDM-staged in LDS) -> iFFT ->
// residual -> SiLU.  The residual x is re-read in the epilogue (guaranteed
// L2-resident: 128 MB tensor < 192 MB L2) instead of being held in registers
// across all WMMA stages, which previously caused scratch spills.
// ===========================================================================
__global__ void s4_conv(const float* __restrict__ x,
                        const float* __restrict__ Dv,
                        const float2* __restrict__ K,
                        float* __restrict__ out) {
    extern __shared__ float smem[];
    float* XR = smem;                       // 9216
    float* XI = XR + 9216;                  // 9216
    float2* KLDS = (float2*)(XI + 9216);    // 8192 float2 = 64 KB
    const int tid = threadIdx.x;
    const int h  = blockIdx.x & 1023;
    const int bb = blockIdx.x >> 10;

#ifdef S4_HAS_TDM
    // --- Tensor Data Mover: async 64KB K[h] row -> KLDS, overlaps fwd FFT ---
    if (tid < 32) {
        unsigned ldsbase = 0;
#if __has_builtin(__builtin_amdgcn_groupstaticsize)
        ldsbase = __builtin_amdgcn_groupstaticsize();
#endif
        unsigned long long ga =
            (unsigned long long)(uintptr_t)(K + (size_t)h * 8192);
        u32x4 g0;
        g0.x = 1u;                                    // count=1 (valid D#)
        g0.y = ldsbase + 2u * 9216u * 4u;             // lds_addr = KLDS offset
        g0.z = (unsigned)ga;                          // global addr [31:0]
        g0.w = (unsigned)((ga >> 32) & 0x01FFFFFFull) | (2u << 30); // +type=2
        i32x8 g1;
        g1[0] = (3 << 16);        // data_size = 8B, no multicast/pad/iterate
        g1[1] = 0x20000000;       // tensor_dim0 = 8192 (bits[63:48])
        g1[2] = 0x00010000;       // tensor_dim1 = 1   (bits[95:80])
        g1[3] = 0x20000000;       // tile_dim0 = 8192  (bits[127:112])
        g1[4] = 1;                // tile_dim1 = 1
        g1[5] = 8192;             // tensor_dim0_stride
        g1[6] = 0;  g1[7] = 0;
        i32x4 g2;  g2[0] = 1; g2[1] = 1; g2[2] = 0; g2[3] = 0;
        i32x4 g3;  g3[0] = 0; g3[1] = 0; g3[2] = 0; g3[3] = 0;
#if __clang_major__ >= 23
        i32x8 g4 = {0, 0, 0, 0, 0, 0, 0, 0};
        __builtin_amdgcn_tensor_load_to_lds(g0, g1, g2, g3, g4, 0);
#else
        __builtin_amdgcn_tensor_load_to_lds(g0, g1, g2, g3, 0);
#endif
    }
#else
    for (int idx = tid; idx < 8192; idx += WG)
        KLDS[idx] = K[(size_t)h * 8192 + idx];
#endif

    // --- S1: strided global load + register FFT-32 over m2 + fused W512 tw ---
    const int a = tid >> 4;     // n1
    const int b = tid & 15;     // m1
    float xr[32], xi[32];
#pragma unroll
    for (int c = 0; c < 16; ++c) {
        int l = a + (b << 4) + (c << 8);
        float v = x[(size_t)l * 8192 + (size_t)bb * 1024 + h];
        xr[c] = v;          xi[c] = 0.0f;
        xr[c + 16] = 0.0f;  xi[c + 16] = 0.0f;
    }
    reg_fft_dif<5>(xr, xi, -1.0f);
#pragma unroll
    for (int j = 0; j < 32; ++j) {
        int k2 = (int)bitrev((unsigned)j, 5);
        float ang = -PI2 * (float)(b * k2) * (1.0f / 512.0f);
        float wr, wi;
        __sincosf(ang, &wi, &wr);
        float t1 = xr[j] * wr - xi[j] * wi;
        xi[j] = xr[j] * wi + xi[j] * wr;
        xr[j] = t1;
    }
    {
        float4* qr = (float4*)(XR + a * AS8 + b * BS8);
        float4* qi = (float4*)(XI + a * AS8 + b * BS8);
#pragma unroll
        for (int q = 0; q < 8; ++q) {
            qr[q] = make_float4(xr[4*q], xr[4*q+1], xr[4*q+2], xr[4*q+3]);
            qi[q] = make_float4(xi[4*q], xi[4*q+1], xi[4*q+2], xi[4*q+3]);
        }
    }
    // --- forward WMMA stages ---
    dft16_wmma(XR, XI, BS8, AS8, 5, 1, -1.0f, 1.0f / 8192.0f);
    dft16_wmma(XR, XI, AS8, BS8, 5, 0, -1.0f, 0.0f);

    // --- wait for TDM, then pointwise multiply with K (same storage order) ---
#ifdef S4_HAS_TDM
    if (tid < 32) {
#if __has_builtin(__builtin_amdgcn_s_wait_tensorcnt)
        __builtin_amdgcn_s_wait_tensorcnt(0);
#else
        asm volatile("s_wait_tensorcnt 0" ::: "memory");
#endif
    }
#endif
    __syncthreads();
    for (int idx = tid; idx < 8192; idx += WG) {
        float2 kv = KLDS[idx];
        int p = (idx >> 9) * AS8 + ((idx >> 5) & 15) * BS8 + (idx & 31);
        float pr = XR[p], pim = XI[p];
        XR[p] = pr * kv.x - pim * kv.y;
        XI[p] = pr * kv.y + pim * kv.x;
    }
    // --- inverse WMMA stages ---
    dft16_wmma(XR, XI, AS8, BS8, 5, 2, +1.0f, 1.0f / 8192.0f);
    dft16_wmma(XR, XI, BS8, AS8, 5, 3, +1.0f, 1.0f / 512.0f);

    // --- I3: register iFFT-32 + scale + residual (x re-read, L2 hit) + SiLU ---
    {
        const float Dh = Dv[h];
        float4* qr = (float4*)(XR + a * AS8 + b * BS8);
        float4* qi = (float4*)(XI + a * AS8 + b * BS8);
#pragma unroll
        for (int q = 0; q < 8; ++q) {
            float4 r4 = qr[q], i4 = qi[q];
            xr[4*q+0] = r4.x; xr[4*q+1] = r4.y; xr[4*q+2] = r4.z; xr[4*q+3] = r4.w;
            xi[4*q+0] = i4.x; xi[4*q+1] = i4.y; xi[4*q+2] = i4.z; xi[4*q+3] = i4.w;
        }
        reg_fft_dit<5>(xr, xi, +1.0f);
#pragma unroll
        for (int c = 0; c < 16; ++c) {
            int l = a + (b << 4) + (c << 8);
            size_t gidx = (size_t)l * 8192 + (size_t)bb * 1024 + h;
            float y = xr[c] * (1.0f / 8192.0f);
            float v = y + x[gidx] * Dh;
            out[gidx] = v * (1.0f / (1.0f + __expf(-v)));
        }
    }
}

// ===========================================================================
extern "C" void kernel_launch(void* const* d_in, const int* in_sizes, int n_in,
                              void* d_out, int out_size, void* d_ws, size_t ws_size,
                              hipStream_t stream) {
    (void)in_sizes; (void)n_in; (void)out_size; (void)ws_size;
    const float* x      = (const float*)d_in[0];
    const float* log_dt = (const float*)d_in[1];
    const float* ivr    = (const float*)d_in[2];
    const float* wim    = (const float*)d_in[3];
    const float* Bp     = (const float*)d_in[4];
    const float* Pp     = (const float*)d_in[5];
    const float* Cp     = (const float*)d_in[6];
    const float* Dv     = (const float*)d_in[7];
    float* out = (float*)d_out;

    char* ws = (char*)d_ws;
    float2* kf = (float2*)ws;                                          // 16.8 MB
    float2* K  = (float2*)(ws + (size_t)1024 * 2049 * sizeof(float2)); // 67.1 MB

    const size_t smem2 = (size_t)(2 * 9216) * sizeof(float);               // 73728
    const size_t smem3 = (size_t)(2 * 9216) * sizeof(float) + 8192 * 8;    // 139264

    s4_cauchy_kernel<<<1024, WG, 0, stream>>>(log_dt, ivr, wim, Bp, Pp, Cp, kf);
    s4_kf_to_K<<<1024, WG, smem2, stream>>>(kf, K);
    s4_conv<<<8192, WG, smem3, stream>>>(x, Dv, K, out);
}